// PitchShift_25469156065333
// MI455X (gfx1250) — compile-verified
//
#include <hip/hip_runtime.h>
#include <math.h>

// ---------------------------------------------------------------------------
// PitchShift pipeline for MI455X (gfx1250), wave32, f32 WMMA 16x16x4.
// STFT / iSTFT as DFT GEMMs on the matrix pipe (f32 for accuracy; problem is
// L2-resident + bandwidth bound, so low-precision WMMA buys nothing).
// ---------------------------------------------------------------------------

typedef __attribute__((ext_vector_type(2))) float v2f;
typedef __attribute__((ext_vector_type(8))) float v8f;

#define BATCH 8
#define TLEN 131072
#define WINL 512
#define HOP 128
#define NF 1025          // frames = T/HOP + 1
#define NFP 1040         // frames padded to multiple of 16
#define NB 257           // rfft bins
#define NBP 272          // bins padded to multiple of 16
#define SLEN 2054        // stretched frames = 2*NF + 4
#define SLENP 2064       // padded to multiple of 16
#define KK2 544          // 272 re + 272 im (inverse GEMM K)
#define TOTAL_OLA 263296 // (SLEN-1)*HOP + WINL
#define PI_F 3.14159265358979323846f
#define TWO_PI_F 6.28318530717958647692f

// workspace layout (float offsets)
#define OFF_WCT ((size_t)0)                               // [NBP][WINL] cos, transposed (freq-major)
#define OFF_WST (OFF_WCT + (size_t)NBP * WINL)            // [NBP][WINL] -sin
#define OFF_MINV (OFF_WST + (size_t)NBP * WINL)           // [WINL][KK2] inverse-DFT * window, transposed
#define OFF_FRAMES (OFF_MINV + (size_t)WINL * KK2)        // [B][NFP][WINL]
#define OFF_SPECRE (OFF_FRAMES + (size_t)BATCH * NFP * WINL) // [B][NFP][NBP]
#define OFF_SPECIM (OFF_SPECRE + (size_t)BATCH * NFP * NBP)
#define OFF_STR (OFF_SPECIM + (size_t)BATCH * NFP * NBP)  // [B][SLENP][KK2]
#define OFF_TFR (OFF_STR + (size_t)BATCH * SLENP * KK2)   // [B][SLENP][WINL]
#define OFF_Y (OFF_TFR + (size_t)BATCH * SLENP * WINL)    // [B][TOTAL_OLA]
#define TOTAL_F (OFF_Y + (size_t)BATCH * TOTAL_OLA)

static __device__ __forceinline__ float hannw(int i) {
    return 0.5f * (1.0f - cosf((TWO_PI_F / (float)WINL) * (float)i));
}

static __device__ __forceinline__ v8f wmma_f32x4(v2f a, v2f b, v8f c) {
    // V_WMMA_F32_16X16X4_F32: D = A(16x4) * B(4x16) + C(16x16)
    return __builtin_amdgcn_wmma_f32_16x16x4_f32(false, a, false, b, (short)0, c,
                                                 false, false);
}

// --- tables: forward cos / -sin (transposed) and windowed inverse DFT --------
__global__ void build_tables(float* __restrict__ ws) {
    int idx = blockIdx.x * blockDim.x + threadIdx.x;
    if (idx >= NB * WINL) return;
    int f = idx / WINL, n = idx % WINL;
    int m = (f * n) & (WINL - 1);                 // exact modular phase
    float ang = (TWO_PI_F / (float)WINL) * (float)m;
    float c = cosf(ang), s = sinf(ang);
    ws[OFF_WCT + (size_t)f * WINL + n] = c;       // Re:  sum x cos
    ws[OFF_WST + (size_t)f * WINL + n] = -s;      // Im: -sum x sin
    // inverse: y[n] = (1/512)[X0 + (-1)^n X256 + 2*sum (Re cos - Im sin)],
    // window folded in.
    float cf = (f == 0 || f == NB - 1) ? 1.0f : 2.0f;
    float sc = (cf / (float)WINL) * hannw(n);
    ws[OFF_MINV + (size_t)n * KK2 + f] = c * sc;
    ws[OFF_MINV + (size_t)n * KK2 + NBP + f] = -s * sc;
}

// --- framing with reflect pad + hann window ---------------------------------
__global__ void frame_window(const float* __restrict__ wav, float* __restrict__ ws) {
    int idx = blockIdx.x * blockDim.x + threadIdx.x;
    if (idx >= BATCH * NF * WINL) return;
    int b = idx / (NF * WINL);
    int rem = idx % (NF * WINL);
    int j = rem / WINL, i = rem % WINL;
    int q = j * HOP + i - (WINL / 2);
    if (q < 0) q = -q;
    else if (q >= TLEN) q = 2 * TLEN - 2 - q;
    ws[OFF_FRAMES + ((size_t)b * NFP + j) * WINL + i] = wav[(size_t)b * TLEN + q] * hannw(i);
}

// --- STFT as GEMM: frames[1040x512] x {cos,-sin}[512x272] via f32 WMMA ------
__global__ __launch_bounds__(32) void stft_wmma(float* __restrict__ ws) {
    const float* frames = ws + OFF_FRAMES;
    const float* wct = ws + OFF_WCT;
    const float* wst = ws + OFF_WST;
    float* sre = ws + OFF_SPECRE;
    float* sim = ws + OFF_SPECIM;
    int fr0 = blockIdx.x * 16, nb0 = blockIdx.y * 16, b = blockIdx.z;
    int lane = threadIdx.x, half = lane >> 4, l = lane & 15;
    // A: lane(0-15)=row M, half selects K pair {0,1}/{2,3}
    const float* arow = frames + ((size_t)(b * NFP + fr0 + l)) * WINL + 2 * half;
    // B stored transposed (freq-major) so the K pair is a contiguous float2
    const float* bcr = wct + (size_t)(nb0 + l) * WINL + 2 * half;
    const float* bsr = wst + (size_t)(nb0 + l) * WINL + 2 * half;
    v8f accR = {}, accI = {};
    for (int k = 0; k < WINL; k += 4) {
        v2f a = *(const v2f*)(arow + k);
        v2f bc = *(const v2f*)(bcr + k);
        v2f bs = *(const v2f*)(bsr + k);
        accR = wmma_f32x4(a, bc, accR);   // shares A tile across both WMMAs
        accI = wmma_f32x4(a, bs, accI);
    }
    for (int r = 0; r < 8; ++r) {
        size_t o = ((size_t)(b * NFP + fr0 + r + 8 * half)) * NBP + nb0 + l;
        sre[o] = accR[r];
        sim[o] = accI[r];
    }
}

// --- phase vocoder: wave32 chunked prefix-scan of wrapped phase -------------
__global__ __launch_bounds__(32) void phase_vocoder(float* __restrict__ ws,
                                                    const int* __restrict__ steps) {
    int bf = blockIdx.x;
    int b = bf / NB, f = bf % NB;
    int lane = threadIdx.x;
    const float* sre = ws + OFF_SPECRE + (size_t)b * NFP * NBP;
    const float* sim = ws + OFF_SPECIM + (size_t)b * NFP * NBP;
    float* str = ws + OFF_STR + (size_t)b * SLENP * KK2;

    float rate = exp2f(-(float)steps[b] / 12.0f);
    int lens = (int)ceilf((float)NF / rate);
    float lastv = (float)(lens - 1) * rate;
    float padv = (float)f * (PI_F * 0.5f);  // linspace(0, pi*HOP, 257)[f]

    float p0r = sre[f], p0i = sim[f];
    float phase0 = (p0r == 0.0f && p0i == 0.0f) ? 0.0f : atan2f(p0i, p0r);
    float carry = phase0;  // acc(t) = phase0 + sum_{v<t} wrapped_ph(v)

    for (int c = 0; c < SLEN; c += 32) {
        int t = c + lane;
        bool valid = t < SLEN;
        float val = valid ? ((t < lens) ? (float)t * rate : lastv) : 0.0f;
        int idx0 = (int)val;
        float alpha = val - (float)idx0;
        float s0r = 0.f, s0i = 0.f, s1r = 0.f, s1i = 0.f;
        if (valid) {
            size_t o0 = (size_t)idx0 * NBP + f;
            s0r = sre[o0]; s0i = sim[o0];
            s1r = sre[o0 + NBP]; s1i = sim[o0 + NBP];
        }
        bool z0 = (s0r == 0.f && s0i == 0.f), z1 = (s1r == 0.f && s1i == 0.f);
        float n0 = z0 ? 0.f : sqrtf(s0r * s0r + s0i * s0i);
        float a0 = z0 ? 0.f : atan2f(s0i, s0r);
        float n1 = z1 ? 0.f : sqrtf(s1r * s1r + s1i * s1i);
        float a1 = z1 ? 0.f : atan2f(s1i, s1r);
        float ph = a1 - a0 - padv;
        ph -= TWO_PI_F * roundf(ph / TWO_PI_F);
        ph += padv;
        float g = valid ? ph : 0.0f;
        // wave32 inclusive scan
        float incl = g;
        for (int off = 1; off < 32; off <<= 1) {
            float nbr = __shfl_up(incl, off, 32);
            if (lane >= off) incl += nbr;
        }
        float acc = carry + (incl - g);          // exclusive prefix
        float total = __shfl(incl, 31, 32);
        if (valid) {
            float mag = alpha * n1 + (1.0f - alpha) * n0;
            float sa, ca;
            sincosf(acc, &sa, &ca);
            size_t ot = (size_t)t * KK2;
            str[ot + f] = mag * ca;
            str[ot + NBP + f] = mag * sa;
        }
        carry += total;
    }
}

// --- iSTFT as GEMM: stretched[2064x544] x Minv[544x512] via f32 WMMA --------
__global__ __launch_bounds__(32) void istft_wmma(float* __restrict__ ws) {
    const float* str = ws + OFF_STR;
    const float* minv = ws + OFF_MINV;
    float* tfr = ws + OFF_TFR;
    int t0 = blockIdx.x * 16, n0 = blockIdx.y * 16, b = blockIdx.z;
    int lane = threadIdx.x, half = lane >> 4, l = lane & 15;
    const float* arow = str + ((size_t)b * SLENP + t0 + l) * KK2 + 2 * half;
    const float* brow = minv + (size_t)(n0 + l) * KK2 + 2 * half;  // transposed
    v8f acc = {};
    for (int k = 0; k < KK2; k += 4) {
        v2f a = *(const v2f*)(arow + k);
        v2f bb = *(const v2f*)(brow + k);
        acc = wmma_f32x4(a, bb, acc);
    }
    for (int r = 0; r < 8; ++r) {
        tfr[((size_t)b * SLENP + t0 + r + 8 * half) * WINL + n0 + l] = acc[r];
    }
}

// --- overlap-add + window-envelope normalization ----------------------------
__global__ void overlap_add(float* __restrict__ ws) {
    int idx = blockIdx.x * blockDim.x + threadIdx.x;
    if (idx >= BATCH * TOTAL_OLA) return;
    int b = idx / TOTAL_OLA, p = idx % TOTAL_OLA;
    int jhi = p / HOP; if (jhi > SLEN - 1) jhi = SLEN - 1;
    int jlo = (p >= WINL) ? ((p - WINL) / HOP + 1) : 0;
    const float* tfr = ws + OFF_TFR;
    float acc = 0.0f, env = 0.0f;
    for (int j = jlo; j <= jhi; ++j) {
        int i = p - j * HOP;
        acc += tfr[((size_t)b * SLENP + j) * WINL + i];
        float w = hannw(i);
        env += w * w;
    }
    ws[OFF_Y + (size_t)idx] = acc / (env > 1e-11f ? env : 1.0f);
}

// --- ragged linear resample back to T + clip --------------------------------
__global__ void interp_out(const float* __restrict__ ws, const int* __restrict__ steps,
                           float* __restrict__ out) {
    int idx = blockIdx.x * blockDim.x + threadIdx.x;
    if (idx >= BATCH * TLEN) return;
    int b = idx / TLEN, t = idx % TLEN;
    float rate = exp2f(-(float)steps[b] / 12.0f);
    int s = (int)floorf((float)TLEN / rate);
    float scale = (float)s / (float)TLEN;
    float src = ((float)t + 0.5f) * scale - 0.5f;
    float smax = (float)(s - 1);
    src = fminf(fmaxf(src, 0.0f), smax);
    int i0 = (int)src;
    int i1 = i0 + 1; if (i1 > s - 1) i1 = s - 1;
    float w = src - (float)i0;
    const float* yb = ws + OFF_Y + (size_t)b * TOTAL_OLA + (WINL / 2); // drop half-window
    float v = (1.0f - w) * yb[i0] + w * yb[i1];
    out[idx] = fminf(fmaxf(v, -1.0f), 1.0f);
}

extern "C" void kernel_launch(void* const* d_in, const int* in_sizes, int n_in,
                              void* d_out, int out_size, void* d_ws, size_t ws_size,
                              hipStream_t stream) {
    (void)in_sizes; (void)n_in; (void)out_size; (void)ws_size;
    const float* wav = (const float*)d_in[0];
    const int* steps = (const int*)d_in[1];
    float* out = (float*)d_out;
    float* ws = (float*)d_ws;

    // zero-init workspace (tables' pad rows, frame/spec/stretch padding)
    hipMemsetAsync(d_ws, 0, TOTAL_F * sizeof(float), stream);

    build_tables<<<(NB * WINL + 255) / 256, 256, 0, stream>>>(ws);
    frame_window<<<(BATCH * NF * WINL + 255) / 256, 256, 0, stream>>>(wav, ws);
    stft_wmma<<<dim3(NFP / 16, NBP / 16, BATCH), 32, 0, stream>>>(ws);
    phase_vocoder<<<dim3(BATCH * NB), 32, 0, stream>>>(ws, steps);
    istft_wmma<<<dim3(SLENP / 16, WINL / 16, BATCH), 32, 0, stream>>>(ws);
    overlap_add<<<(BATCH * TOTAL_OLA + 255) / 256, 256, 0, stream>>>(ws);
    interp_out<<<(BATCH * TLEN + 255) / 256, 256, 0, stream>>>(ws, steps, out);
}